// DenseToSparse_82437602279968
// MI455X (gfx1250) — compile-verified
//
#include <hip/hip_runtime.h>
#include <hip/hip_bf16.h>
#include <stdint.h>

// Problem geometry (fixed by the reference)
#define B_   32
#define C_   256
#define H_   64
#define W_   64
#define HW_  (H_ * W_)            // 4096
#define NSITES (B_ * HW_)         // 131072
#define GROUP  32                 // sites per compaction group (one wave-width)
#define NGROUPS (NSITES / GROUP)  // 4096
#define ROWSTRIDE 36              // LDS floats per channel row (32 data + 4 pad; 144B, 16B aligned)

typedef float f32x4 __attribute__((ext_vector_type(4)));
typedef int   i32x4 __attribute__((ext_vector_type(4)));

#ifndef __has_builtin
#define __has_builtin(x) 0
#endif

#if __has_builtin(__builtin_amdgcn_global_load_async_to_lds_b128)
#define HAVE_ASYNC_LDS 1
// builtin signature (probe-confirmed via round-1 diagnostic):
//   (int4 addrspace(1)* gsrc, int4 addrspace(3)* ldst, imm offset, imm cpol)
typedef __attribute__((address_space(1))) i32x4 gv4_t;
typedef __attribute__((address_space(3))) i32x4 lv4_t;
#else
#define HAVE_ASYNC_LDS 0
#endif

__device__ __forceinline__ void wait_async_zero() {
#if __has_builtin(__builtin_amdgcn_s_wait_asynccnt)
    __builtin_amdgcn_s_wait_asynccnt(0);
#else
    asm volatile("s_wait_asynccnt 0" ::: "memory");
#endif
}

// ---------------------------------------------------------------------------
// Kernel 1: per-site active flags + per-32-site-group counts.
// Thread handles 4 consecutive sites via float4 loads: lanes read 512B
// contiguous per channel iteration -> fully coalesced.
// ---------------------------------------------------------------------------
__global__ void k_flags(const float* __restrict__ x,
                        uint8_t* __restrict__ flags,
                        uint32_t* __restrict__ group_counts) {
    __shared__ uint32_t shc[256];
    const int tid   = threadIdx.x;
    const int site4 = (blockIdx.x * 256 + tid) * 4;     // first of 4 sites
    const int b     = site4 >> 12;                       // / HW_
    const int hw    = site4 & (HW_ - 1);
    const float* base = x + (size_t)b * C_ * HW_ + hw;

    int nz0 = 0, nz1 = 0, nz2 = 0, nz3 = 0;
    #pragma unroll 8
    for (int c = 0; c < C_; ++c) {
        f32x4 v = *(const f32x4*)(base + (size_t)c * HW_);
        nz0 |= (v.x != 0.0f);
        nz1 |= (v.y != 0.0f);
        nz2 |= (v.z != 0.0f);
        nz3 |= (v.w != 0.0f);
    }
    const uint32_t f0 = nz0 ? 1u : 0u, f1 = nz1 ? 1u : 0u;
    const uint32_t f2 = nz2 ? 1u : 0u, f3 = nz3 ? 1u : 0u;
    *(uint32_t*)(flags + site4) = f0 | (f1 << 8) | (f2 << 16) | (f3 << 24);

    shc[tid] = f0 + f1 + f2 + f3;
    __syncthreads();
    if (tid < 32) {                     // 32 groups of 32 sites per block
        uint32_t s = 0;
        #pragma unroll
        for (int i = 0; i < 8; ++i) s += shc[tid * 8 + i];
        group_counts[blockIdx.x * 32 + tid] = s;
    }
}

// ---------------------------------------------------------------------------
// Kernel 2: exclusive scan of 4096 group counts (single 1024-thread block).
// ---------------------------------------------------------------------------
__global__ void k_scan(const uint32_t* __restrict__ gcnt,
                       uint32_t* __restrict__ goff,
                       uint32_t* __restrict__ total) {
    __shared__ uint32_t sh[1024];
    const int tid = threadIdx.x;
    const uint32_t c0 = gcnt[tid * 4 + 0];
    const uint32_t c1 = gcnt[tid * 4 + 1];
    const uint32_t c2 = gcnt[tid * 4 + 2];
    const uint32_t c3 = gcnt[tid * 4 + 3];
    const uint32_t t  = c0 + c1 + c2 + c3;
    sh[tid] = t;
    __syncthreads();
    for (int off = 1; off < 1024; off <<= 1) {
        uint32_t v = 0;
        if (tid >= off) v = sh[tid - off];
        __syncthreads();
        sh[tid] += v;
        __syncthreads();
    }
    const uint32_t excl = sh[tid] - t;
    goff[tid * 4 + 0] = excl;
    goff[tid * 4 + 1] = excl + c0;
    goff[tid * 4 + 2] = excl + c0 + c1;
    goff[tid * 4 + 3] = excl + c0 + c1 + c2;
    if (tid == 1023) total[0] = sh[1023];
}

// ---------------------------------------------------------------------------
// Kernel 3: compaction. One block per 32-site group, 256 threads.
// The 32x256 transpose tile is DMA'd global->LDS with async b128 copies,
// issued FIRST so the flag fetch + 32-wide prefix scan overlap the DMA.
// Then active rows are written thread-per-channel (coalesced 1KB stores,
// consecutive output rows thanks to the stable scan).
// ---------------------------------------------------------------------------
__global__ void k_compact(const float* __restrict__ x,
                          const uint8_t* __restrict__ flags,
                          const uint32_t* __restrict__ goff,
                          float* __restrict__ out) {
    __shared__ float tile[C_ * ROWSTRIDE];   // tile[c*36 + s]
    __shared__ uint32_t sh_flag[GROUP];
    __shared__ uint32_t sh_list[GROUP];
    __shared__ uint32_t sh_k;

    const int tid = threadIdx.x;
    const int g   = blockIdx.x;
    const int n0  = g * GROUP;
    const int b   = n0 >> 12;
    const int hw0 = n0 & (HW_ - 1);
    const float* gbase = x + (size_t)b * C_ * HW_ + hw0;

    // --- issue tile DMA first: 32 sites x 256 ch = 2048 b128 chunks, 8/thread
#if HAVE_ASYNC_LDS
    #pragma unroll
    for (int it = 0; it < 8; ++it) {
        const int q = it * 256 + tid;        // 0..2047
        const int c = q >> 3;                // channel
        const int k = q & 7;                 // 16B chunk within the 32-site row
        const float* gsrc = gbase + (size_t)c * HW_ + k * 4;
        float* ldst = &tile[c * ROWSTRIDE + k * 4];
        __builtin_amdgcn_global_load_async_to_lds_b128(
            (gv4_t*)gsrc, (lv4_t*)ldst, 0, 0);
    }
#else
    #pragma unroll
    for (int it = 0; it < 8; ++it) {
        const int q = it * 256 + tid;
        const int c = q >> 3;
        const int k = q & 7;
        f32x4 v = *(const f32x4*)(gbase + (size_t)c * HW_ + k * 4);
        *(f32x4*)&tile[c * ROWSTRIDE + k * 4] = v;
    }
#endif

    // --- overlap: flags + stable intra-group compaction list while DMA flies
    if (tid < GROUP) sh_flag[tid] = flags[n0 + tid];
    __syncthreads();
    if (tid < GROUP) {
        uint32_t pos = 0;
        for (int j = 0; j < tid; ++j) pos += sh_flag[j];
        if (sh_flag[tid]) sh_list[pos] = (uint32_t)tid;
        if (tid == GROUP - 1) sh_k = pos + sh_flag[GROUP - 1];
    }

#if HAVE_ASYNC_LDS
    wait_async_zero();       // per-wave DMA completion
#endif
    __syncthreads();         // all waves' tile portions + sh_list visible

    const uint32_t kact = sh_k;
    const uint32_t base = goff[g];
    for (uint32_t j = 0; j < kact; ++j) {
        const uint32_t s = sh_list[j];
        // thread tid supplies channel tid of output row (base + j)
        out[((size_t)(base + j)) * C_ + tid] = tile[tid * ROWSTRIDE + s];
    }
}

// ---------------------------------------------------------------------------
// Kernel 4: zero the padded tail rows [count, NSITES) with float4 stores.
// ---------------------------------------------------------------------------
__global__ void k_zerotail(float* __restrict__ out,
                           const uint32_t* __restrict__ total) {
    const uint32_t count = *total;
    const size_t startv = (size_t)count * (C_ / 4);
    const size_t endv   = (size_t)NSITES * (C_ / 4);
    f32x4 z = {0.0f, 0.0f, 0.0f, 0.0f};
    f32x4* out4 = (f32x4*)out;
    for (size_t i = startv + (size_t)blockIdx.x * blockDim.x + threadIdx.x;
         i < endv; i += (size_t)gridDim.x * blockDim.x) {
        out4[i] = z;
    }
}

// ---------------------------------------------------------------------------
extern "C" void kernel_launch(void* const* d_in, const int* in_sizes, int n_in,
                              void* d_out, int out_size, void* d_ws, size_t ws_size,
                              hipStream_t stream) {
    (void)in_sizes; (void)n_in; (void)out_size; (void)ws_size;
    const float* x = (const float*)d_in[0];
    float* out = (float*)d_out;

    uint8_t* ws = (uint8_t*)d_ws;
    uint32_t* total = (uint32_t*)ws;                              // 4B (16B slot)
    uint32_t* gcnt  = (uint32_t*)(ws + 16);                       // 4096 u32
    uint32_t* goff  = (uint32_t*)(ws + 16 + NGROUPS * 4);         // 4096 u32
    uint8_t*  flags = ws + 16 + NGROUPS * 8;                      // NSITES bytes

    k_flags   <<<NSITES / 1024, 256, 0, stream>>>(x, flags, gcnt);
    k_scan    <<<1, 1024, 0, stream>>>(gcnt, goff, total);
    k_compact <<<NGROUPS, 256, 0, stream>>>(x, flags, goff, out);
    k_zerotail<<<1024, 256, 0, stream>>>(out, total);
}